// SelfAttentionConv2d_77962246357255
// MI455X (gfx1250) — compile-verified
//
#include <hip/hip_runtime.h>

// Self-attention conv2d (7x7 window, per-channel softmax), MI455X / gfx1250.
// Phase 1: q/k/v 1x1 projections via V_WMMA_F32_16X16X4_F32 (fp32-exact).
// Phase 2: per-(b,f)-plane windowed softmax with LDS-staged k/v planes.

typedef __attribute__((ext_vector_type(2))) float v2f;
typedef __attribute__((ext_vector_type(8))) float v8f;

#define B_  4
#define C_  256
#define F_  256
#define H_  32
#define W_  32
#define HW_ 1024           // 32*32
#define PLANE_ELEMS (B_ * F_ * HW_)   // 1,048,576 floats per tensor

// ---------------------------------------------------------------------------
// Phase 1: Q/K/V projection GEMM.
//   q[b,f,p] = sum_c x[b,c,p] * Wq[c,f]   (same for k, v)
// One wave per 16(M=pixels) x 16(N=f) tile. A tile loaded once per K-step and
// reused for all three B matrices -> 3 WMMAs per 4-deep K-step, 64 steps.
// ---------------------------------------------------------------------------
__global__ __launch_bounds__(32) void proj_qkv_kernel(
    const float* __restrict__ x,
    const float* __restrict__ Wq, const float* __restrict__ Wk,
    const float* __restrict__ Wv,
    float* __restrict__ qo, float* __restrict__ ko, float* __restrict__ vo)
{
    const int lane = threadIdx.x;       // 0..31, wave32
    const int half = lane >> 4;         // 0 or 1
    const int ml   = lane & 15;         // M (A) / N (B,C) lane index

    const int m0   = blockIdx.x * 16;   // global pixel-row of tile
    const int bidx = m0 >> 10;          // batch (1024 pixels per batch plane)
    const int hw0  = m0 & 1023;
    const int n0   = blockIdx.y * 16;   // output-feature column of tile

    // A element (m, k) lives at x + bidx*C_*HW_ + k*HW_ + (hw0 + m)
    const float* xb = x + (size_t)bidx * (C_ * HW_) + hw0 + ml;

    v8f cq = {}; v8f ck = {}; v8f cv = {};

    for (int k0 = 0; k0 < C_; k0 += 4) {
        const int ka = k0 + 2 * half;   // this half-wave's K pair base
        // A 16x4 fp32: v0 = K{0,2}, v1 = K{1,3} split across lane halves
        v2f a;
        a.x = xb[(size_t)(ka + 0) * HW_];
        a.y = xb[(size_t)(ka + 1) * HW_];
        // B 4x16 fp32: lanes hold N, v0/v1 hold the same K pair
        const float* wq = Wq + (size_t)ka * F_ + n0 + ml;
        const float* wk = Wk + (size_t)ka * F_ + n0 + ml;
        const float* wv = Wv + (size_t)ka * F_ + n0 + ml;
        v2f bq; bq.x = wq[0]; bq.y = wq[F_];
        v2f bk; bk.x = wk[0]; bk.y = wk[F_];
        v2f bv; bv.x = wv[0]; bv.y = wv[F_];

        cq = __builtin_amdgcn_wmma_f32_16x16x4_f32(false, a, false, bq,
                                                   (short)0, cq, false, false);
        ck = __builtin_amdgcn_wmma_f32_16x16x4_f32(false, a, false, bk,
                                                   (short)0, ck, false, false);
        cv = __builtin_amdgcn_wmma_f32_16x16x4_f32(false, a, false, bv,
                                                   (short)0, cv, false, false);
    }

    // C/D layout: lane -> N = n0+ml, vgpr i -> M = half*8 + i.
    // Per lane the 8 values land at consecutive hw addresses (b128-mergeable).
    const size_t obase = (size_t)bidx * (F_ * HW_) +
                         (size_t)(n0 + ml) * HW_ + hw0 + half * 8;
    float* qp = qo + obase;
    float* kp = ko + obase;
    float* vp = vo + obase;
#pragma unroll
    for (int i = 0; i < 8; ++i) {
        qp[i] = cq[i];
        kp[i] = ck[i];
        vp[i] = cv[i];
    }
}

// ---------------------------------------------------------------------------
// Phase 2: windowed per-channel softmax attention.
// One block per (b,f) plane. k/v planes staged in LDS (2 x 4KB).
// rel[ky,kx] for channel f is rank-1: f<128 -> rel_x[kx,f], else rel_y[ky,f-128]
// Padding semantics: OOB window position has k=0 (logit = q*rel, still in the
// softmax denominator) and v=0 (no numerator contribution).
// ---------------------------------------------------------------------------
__global__ __launch_bounds__(256) void attn_kernel(
    const float* __restrict__ q, const float* __restrict__ k,
    const float* __restrict__ v,
    const float* __restrict__ rel_x, const float* __restrict__ rel_y,
    const float* __restrict__ bias, float* __restrict__ out)
{
    __shared__ float sK[HW_];
    __shared__ float sV[HW_];
    __shared__ float sRel[7];
    __shared__ float sBias;

    const int tid = threadIdx.x;
    const int pf  = blockIdx.x;          // b*256 + f
    const int f   = pf & (F_ - 1);
    const size_t base = (size_t)pf * HW_;

#pragma unroll
    for (int it = 0; it < 4; ++it) {
        const int idx = tid + it * 256;
        sK[idx] = k[base + idx];
        sV[idx] = v[base + idx];
    }
    if (tid < 7)
        sRel[tid] = (f < 128) ? rel_x[tid * 128 + f]
                              : rel_y[tid * 128 + (f - 128)];
    if (tid == 8) sBias = bias[f];
    __syncthreads();

    const bool use_kx = (f < 128);

#pragma unroll
    for (int it = 0; it < 4; ++it) {
        const int p = tid + it * 256;
        const int h = p >> 5;
        const int w = p & 31;
        const float qc = q[base + p];

        // pass 1: running max of logits
        float mx = -3.402823466e38f;
        for (int ky = 0; ky < 7; ++ky) {
            const int hh = h + ky - 3;
            const bool hin = ((unsigned)hh < (unsigned)H_);
            const float ry = sRel[ky];
#pragma unroll
            for (int kx = 0; kx < 7; ++kx) {
                const int ww = w + kx - 3;
                const bool in = hin && ((unsigned)ww < (unsigned)W_);
                const float kv = in ? sK[hh * W_ + ww] : 0.0f;
                const float r  = use_kx ? sRel[kx] : ry;
                mx = fmaxf(mx, qc * (kv + r));
            }
        }

        // pass 2: exp-sum and weighted v accumulation
        float den = 0.0f, num = 0.0f;
        for (int ky = 0; ky < 7; ++ky) {
            const int hh = h + ky - 3;
            const bool hin = ((unsigned)hh < (unsigned)H_);
            const float ry = sRel[ky];
#pragma unroll
            for (int kx = 0; kx < 7; ++kx) {
                const int ww = w + kx - 3;
                const bool in = hin && ((unsigned)ww < (unsigned)W_);
                const float kv = in ? sK[hh * W_ + ww] : 0.0f;
                const float vv = in ? sV[hh * W_ + ww] : 0.0f;
                const float r  = use_kx ? sRel[kx] : ry;
                const float e  = __expf(qc * (kv + r) - mx);
                den += e;
                num = fmaf(e, vv, num);
            }
        }
        out[base + p] = num / den + sBias;
    }
}

// ---------------------------------------------------------------------------
extern "C" void kernel_launch(void* const* d_in, const int* in_sizes, int n_in,
                              void* d_out, int out_size, void* d_ws,
                              size_t ws_size, hipStream_t stream)
{
    (void)in_sizes; (void)n_in; (void)out_size; (void)ws_size;

    const float* x     = (const float*)d_in[0];
    const float* Wq    = (const float*)d_in[1];
    const float* Wk    = (const float*)d_in[2];
    const float* Wv    = (const float*)d_in[3];
    const float* rel_x = (const float*)d_in[4];
    const float* rel_y = (const float*)d_in[5];
    const float* bias  = (const float*)d_in[6];
    float* out = (float*)d_out;

    // workspace: q, k, v planes (4 MB each, L2-resident)
    float* qw = (float*)d_ws;
    float* kw = qw + PLANE_ELEMS;
    float* vw = kw + PLANE_ELEMS;

    // Phase 1: 256 M-tiles x 16 N-tiles, one wave32 per tile.
    dim3 g1(PLANE_ELEMS / (16 * F_) * 4, F_ / 16);  // (1024, 16); unused
    proj_qkv_kernel<<<dim3(256, 16), 32, 0, stream>>>(x, Wq, Wk, Wv, qw, kw, vw);

    // Phase 2: one block per (b,f) plane.
    attn_kernel<<<B_ * F_, 256, 0, stream>>>(qw, kw, vw, rel_x, rel_y, bias, out);
}